// GlowTTS_51161650430158
// MI455X (gfx1250) — compile-verified
//
#include <hip/hip_runtime.h>
#include <math.h>

typedef float v2f __attribute__((ext_vector_type(2)));
typedef float v8f __attribute__((ext_vector_type(8)));

static constexpr int   B      = 32;
static constexpr int   C      = 80;
static constexpr int   K2     = 2 * C;     // 160, stacked [m*p ; -0.5*p] vs [x ; x^2]
static constexpr int   KP     = K2 / 2;    // 80 K-pairs
static constexpr int   TX     = 512;
static constexpr int   TY     = 2048;
static constexpr float NEGV   = -1000000000.0f;
static constexpr float LOG2PI = 1.8378770664093453f;

// ---------------------------------------------------------------------------
// Prep: X2p[b][k/2][ty][2] = {x ; x^2}  (paired-K so WMMA fragments are b64 loads)
// ---------------------------------------------------------------------------
__global__ void prep_x2(const float* __restrict__ x, float* __restrict__ X2p) {
  int i = blockIdx.x * blockDim.x + threadIdx.x;          // over B*C*TY
  if (i >= B * C * TY) return;
  int ty = i % TY;
  int c  = (i / TY) % C;
  int b  = i / (TY * C);
  float v = x[i];
  int k0 = c, k1 = C + c;                                 // (C even -> parity = c&1)
  X2p[(((size_t)b * KP + (k0 >> 1)) * TY + ty) * 2 + (k0 & 1)] = v;
  X2p[(((size_t)b * KP + (k1 >> 1)) * TY + ty) * 2 + (k1 & 1)] = v * v;
}

// ---------------------------------------------------------------------------
// Prep: Pp[b][k/2][tx][2] = {m*p ; -0.5*p},
//       constv[b,tx] = sum_c(-s - 0.5 m^2 p) - C/2*log2pi
// ---------------------------------------------------------------------------
__global__ void prep_p(const float* __restrict__ m, const float* __restrict__ s,
                       float* __restrict__ Pp, float* __restrict__ constv) {
  int i = blockIdx.x * blockDim.x + threadIdx.x;          // over B*TX
  if (i >= B * TX) return;
  int tx = i % TX;
  int b  = i / TX;
  float acc = -0.5f * (float)C * LOG2PI;
  for (int c = 0; c < C; ++c) {
    float mv = m[((size_t)b * C + c) * TX + tx];
    float sv = s[((size_t)b * C + c) * TX + tx];
    float p  = expf(-2.0f * sv);
    int k0 = c, k1 = C + c;
    Pp[(((size_t)b * KP + (k0 >> 1)) * TX + tx) * 2 + (k0 & 1)] = mv * p;
    Pp[(((size_t)b * KP + (k1 >> 1)) * TX + tx) * 2 + (k1 & 1)] = -0.5f * p;
    acc += -sv - 0.5f * mv * mv * p;
  }
  constv[i] = acc;
}

// ---------------------------------------------------------------------------
// Scores GEMM: S[b, ty, tx] = const[b,tx] + sum_k X2[b,k,ty] * P[b,k,tx]
// One wave per 16(M=ty) x 64(N=tx) block: A fragment reused across 4 B tiles,
// fragments loaded as b64 (paired-K layout). fp32 WMMA, K-step 4.
// ---------------------------------------------------------------------------
__global__ void scores_gemm(const float* __restrict__ X2p, const float* __restrict__ Pp,
                            const float* __restrict__ constv, float* __restrict__ S) {
  const int lane = threadIdx.x & 31;
  const int wave = blockIdx.x * (blockDim.x >> 5) + (threadIdx.x >> 5);
  const int NT = TX / 64, MT = TY / 16;                   // 8 x 128 blocks per batch
  const int nt = wave % NT;
  const int mt = (wave / NT) % MT;
  const int b  = wave / (NT * MT);
  if (b >= B) return;
  const int  ty0 = mt * 16, tx0 = nt * 64;
  const int  l15 = lane & 15;
  const bool hi  = lane >= 16;

  v8f acc0 = {}, acc1 = {}, acc2 = {}, acc3 = {};

  // A 16x4 f32 fragment: VGPR0 -> K (lanes 0-15) / K+2 (lanes 16-31); VGPR1 -> K+1 / K+3
  const float* Ab = X2p + ((size_t)b * KP) * TY * 2 + (size_t)(ty0 + l15) * 2;
  const float* Bb = Pp  + ((size_t)b * KP) * TX * 2 + (size_t)(tx0 + l15) * 2;
#pragma unroll 4
  for (int k0 = 0; k0 < K2; k0 += 4) {
    const int kp = (k0 + (hi ? 2 : 0)) >> 1;              // even K -> pair index
    v2f a  = *(const v2f*)(Ab + (size_t)kp * TY * 2);
    v2f b0 = *(const v2f*)(Bb + (size_t)kp * TX * 2);
    v2f b1 = *(const v2f*)(Bb + (size_t)kp * TX * 2 + 32);
    v2f b2 = *(const v2f*)(Bb + (size_t)kp * TX * 2 + 64);
    v2f b3 = *(const v2f*)(Bb + (size_t)kp * TX * 2 + 96);
    acc0 = __builtin_amdgcn_wmma_f32_16x16x4_f32(false, a, false, b0, (short)0, acc0, false, false);
    acc1 = __builtin_amdgcn_wmma_f32_16x16x4_f32(false, a, false, b1, (short)0, acc1, false, false);
    acc2 = __builtin_amdgcn_wmma_f32_16x16x4_f32(false, a, false, b2, (short)0, acc2, false, false);
    acc3 = __builtin_amdgcn_wmma_f32_16x16x4_f32(false, a, false, b3, (short)0, acc3, false, false);
  }

  // D 16x16 f32: VGPR r -> M=r (lanes 0-15) / M=r+8 (lanes 16-31), N = lane&15
  v8f accs[4] = {acc0, acc1, acc2, acc3};
#pragma unroll
  for (int n = 0; n < 4; ++n) {
    const float cadd = constv[b * TX + tx0 + n * 16 + l15];
    float* D = S + ((size_t)b * TY + ty0) * TX + tx0 + n * 16 + l15;
#pragma unroll
    for (int r = 0; r < 8; ++r) {
      int mrow = r + (hi ? 8 : 0);
      D[(size_t)mrow * TX] = accs[n][r] + cadd;
    }
  }
}

// ---------------------------------------------------------------------------
// MAS forward scan, in place: S[b,t,:] becomes Q[b,t,:].
// One wave per batch; lane owns 16 consecutive tx in registers; only the
// shift-by-1 boundary crosses lanes (one __shfl_up per step, no LDS barrier).
// ---------------------------------------------------------------------------
__global__ void mas_forward(float* __restrict__ S, const int* __restrict__ text_len) {
  const int b    = blockIdx.x;
  const int lane = threadIdx.x;                 // 0..31 (wave32)
  const int tl   = text_len[b];
  float q[16];
#pragma unroll
  for (int i = 0; i < 16; ++i) q[i] = NEGV;

  float* rowBase = S + (size_t)b * TY * TX + lane * 16;
#pragma unroll 1
  for (int t = 0; t < TY; ++t) {
    float* row = rowBase + (size_t)t * TX;
    if (t + 1 < TY) __builtin_prefetch(row + TX, 0, 0);
    float4 s0 = *(const float4*)(row);
    float4 s1 = *(const float4*)(row + 4);
    float4 s2 = *(const float4*)(row + 8);
    float4 s3 = *(const float4*)(row + 12);
    float sv[16] = {s0.x, s0.y, s0.z, s0.w, s1.x, s1.y, s1.z, s1.w,
                    s2.x, s2.y, s2.z, s2.w, s3.x, s3.y, s3.z, s3.w};
#pragma unroll
    for (int i = 0; i < 16; ++i)
      if (lane * 16 + i >= tl) sv[i] = NEGV;    // mask tx >= text_length

    float up = __shfl_up(q[15], 1, 32);
    if (lane == 0) up = (t == 0) ? 0.0f : NEGV; // shifted[0] = first
#pragma unroll
    for (int i = 15; i >= 0; --i) {
      float sh = (i == 0) ? up : q[i - 1];
      q[i] = sv[i] + fmaxf(q[i], sh);
    }
    *(float4*)(row)      = make_float4(q[0],  q[1],  q[2],  q[3]);
    *(float4*)(row + 4)  = make_float4(q[4],  q[5],  q[6],  q[7]);
    *(float4*)(row + 8)  = make_float4(q[8],  q[9],  q[10], q[11]);
    *(float4*)(row + 12) = make_float4(q[12], q[13], q[14], q[15]);
  }
}

// ---------------------------------------------------------------------------
// Backtrack: one wave, lane = batch. Per-lane gather of Q[b,t-1,{idx-1,idx}].
// ---------------------------------------------------------------------------
__global__ void mas_backtrack(const float* __restrict__ S, const int* __restrict__ text_len,
                              const int* __restrict__ mels_len, int* __restrict__ idx_t) {
  int b = threadIdx.x;
  if (b >= B) return;
  const int ml = mels_len[b];
  int idx = text_len[b] - 1;
#pragma unroll 1
  for (int t = TY - 1; t >= 0; --t) {
    idx_t[b * TY + t] = idx;
    bool active = t < ml;
    int  tm1 = (t > 0) ? (t - 1) : 0;
    const float* qrow = S + ((size_t)b * TY + tm1) * TX;
    float q_cur = qrow[idx];
    float q_up  = qrow[(idx > 0) ? (idx - 1) : 0];
    bool move = (idx > 0) && ((idx == t) || (q_cur < q_up)) && active && (t > 0);
    if (move) --idx;
  }
}

// ---------------------------------------------------------------------------
// Outputs: attn is one-hot -> the bct,bts->bcs einsums are gathers.
// ---------------------------------------------------------------------------
__global__ void out_means(const float* __restrict__ tm, const float* __restrict__ ts,
                          const int* __restrict__ idx_t, const int* __restrict__ mels_len,
                          float* __restrict__ omean, float* __restrict__ ols) {
  int i = blockIdx.x * blockDim.x + threadIdx.x;          // over B*C*TY
  if (i >= B * C * TY) return;
  int t = i % TY;
  int c = (i / TY) % C;
  int b = i / (TY * C);
  bool active = t < mels_len[b];
  int  j = idx_t[b * TY + t];
  omean[i] = active ? tm[((size_t)b * C + c) * TX + j] : 0.0f;
  ols[i]   = active ? ts[((size_t)b * C + c) * TX + j] : 0.0f;
}

__global__ void out_attn(const int* __restrict__ idx_t, const int* __restrict__ mels_len,
                         float* __restrict__ oat) {
  int i = blockIdx.x * blockDim.x + threadIdx.x;          // over B*TX*TY
  if (i >= B * TX * TY) return;
  int t  = i % TY;
  int tx = (i / TY) % TX;
  int b  = i / (TY * TX);
  bool hit = (t < mels_len[b]) && (tx == idx_t[b * TY + t]);
  oat[i] = hit ? 1.0f : 0.0f;
}

// ---------------------------------------------------------------------------
extern "C" void kernel_launch(void* const* d_in, const int* in_sizes, int n_in,
                              void* d_out, int out_size, void* d_ws, size_t ws_size,
                              hipStream_t stream) {
  (void)in_sizes; (void)n_in; (void)out_size; (void)ws_size;
  const float* x   = (const float*)d_in[0];   // mels_latent    (B,C,Ty)
  const float* tm  = (const float*)d_in[1];   // text_mean      (B,C,Tx)
  const float* ts  = (const float*)d_in[2];   // text_log_stdev (B,C,Tx)
  const int*   tlL = (const int*)d_in[3];     // text_lengths   (B,)
  const int*   mlL = (const int*)d_in[4];     // mels_lengths   (B,)

  // Workspace carve-up
  char*  ws   = (char*)d_ws;
  float* Sbuf = (float*)ws;                           // (B,Ty,Tx)  128 MiB (scores -> Q in place)
  ws += (size_t)B * TY * TX * sizeof(float);
  float* X2p  = (float*)ws;                           // (B,KP,Ty,2) 40 MiB
  ws += (size_t)B * K2 * TY * sizeof(float);
  float* Pp   = (float*)ws;                           // (B,KP,Tx,2) 10 MiB
  ws += (size_t)B * K2 * TX * sizeof(float);
  float* cv   = (float*)ws;                           // (B,Tx)
  ws += (size_t)B * TX * sizeof(float);
  int*   idxT = (int*)ws;                             // (B,Ty)

  float* omean = (float*)d_out;                       // (B,C,Ty)
  float* ols   = omean + (size_t)B * C * TY;          // (B,C,Ty)
  float* oat   = ols   + (size_t)B * C * TY;          // (B,Tx,Ty)

  prep_x2<<<(B * C * TY + 255) / 256, 256, 0, stream>>>(x, X2p);
  prep_p<<<(B * TX + 255) / 256, 256, 0, stream>>>(tm, ts, Pp, cv);

  const int totalWaves = B * (TY / 16) * (TX / 64);   // 32768 waves, 16x64 per wave
  scores_gemm<<<totalWaves / 8, 256, 0, stream>>>(X2p, Pp, cv, Sbuf);

  mas_forward<<<B, 32, 0, stream>>>(Sbuf, tlL);
  mas_backtrack<<<1, 32, 0, stream>>>(Sbuf, tlL, mlL, idxT);

  out_means<<<(B * C * TY + 255) / 256, 256, 0, stream>>>(tm, ts, idxT, mlL, omean, ols);
  out_attn<<<(B * TX * TY + 255) / 256, 256, 0, stream>>>(idxT, mlL, oat);
}